// InfoNCELoss_6648609374967
// MI455X (gfx1250) — compile-verified
//
#include <hip/hip_runtime.h>
#include <math.h>

// ---------------------------------------------------------------------------
// InfoNCE loss for MI455X (gfx1250, wave32).
// Strategy: gather-only formulation (never materialize the 2048x100000 sim
// matrix). One wave per query. Dot products computed on the f32 WMMA pipe
// (V_WMMA_F32_16X16X4_F32) to keep full fp32 precision: A = 16 gathered ref
// rows x 4 dims, B = query vector replicated across all 16 columns.
// Norm folding: sim = dot(r,q) / (||r||*||q||*T).
// ---------------------------------------------------------------------------

typedef __attribute__((ext_vector_type(2))) float v2f;
typedef __attribute__((ext_vector_type(8))) float v8f;

#define DIM        128
#define NCAND      64          // 1 positive + 63 negatives
#define INV_TEMP   (1.0f / 0.07f)
#define EPSN       1e-12f

__device__ __forceinline__ float wave_max32(float v) {
    #pragma unroll
    for (int off = 16; off > 0; off >>= 1)
        v = fmaxf(v, __shfl_xor(v, off, 32));
    return v;
}
__device__ __forceinline__ float wave_sum32(float v) {
    #pragma unroll
    for (int off = 16; off > 0; off >>= 1)
        v += __shfl_xor(v, off, 32);
    return v;
}

__global__ void __launch_bounds__(32)
infonce_per_query_kernel(const float* __restrict__ q_embs,   // [B,128]
                         const float* __restrict__ r_embs,   // [V,128]
                         const int*   __restrict__ pos_idx,  // [B]
                         const int*   __restrict__ neg_idx,  // [B,63]
                         float*       __restrict__ per_loss, // [B]
                         int nneg)
{
    const int b    = blockIdx.x;
    const int lane = threadIdx.x;          // 0..31, wave32

    __shared__ float lds_q[DIM];
    __shared__ float lds_innorm[NCAND];
    __shared__ int   lds_idx[NCAND];
    __shared__ float lds_dot[NCAND];

    // ---- 1. candidate indices: slot 0 = positive, slots 1..63 = negatives
    const int j0 = lane;
    const int j1 = lane + 32;
    const int i0 = (j0 == 0) ? pos_idx[b] : neg_idx[(size_t)b * nneg + (j0 - 1)];
    const int i1 = neg_idx[(size_t)b * nneg + (j1 - 1)];
    lds_idx[j0] = i0;
    lds_idx[j1] = i1;

    // ---- 2. query: load 4 floats/lane, stage to LDS, compute inv-norm
    const float4 qv = *(const float4*)(q_embs + (size_t)b * DIM + lane * 4);
    *(float4*)(&lds_q[lane * 4]) = qv;
    float qss = qv.x * qv.x + qv.y * qv.y + qv.z * qv.z + qv.w * qv.w;
    qss = wave_sum32(qss);
    const float inv_qn = 1.0f / fmaxf(sqrtf(qss), EPSN);

    // ---- 3. gathered-ref inverse norms: 2 rows per lane, serial ssq
    {
        const float* rp0 = r_embs + (size_t)i0 * DIM;
        const float* rp1 = r_embs + (size_t)i1 * DIM;
        float s0 = 0.0f, s1 = 0.0f;
        #pragma unroll 8
        for (int k = 0; k < DIM / 4; ++k) {
            const float4 a = *(const float4*)(rp0 + 4 * k);
            const float4 c = *(const float4*)(rp1 + 4 * k);
            s0 += a.x * a.x + a.y * a.y + a.z * a.z + a.w * a.w;
            s1 += c.x * c.x + c.y * c.y + c.z * c.z + c.w * c.w;
        }
        lds_innorm[j0] = 1.0f / fmaxf(sqrtf(s0), EPSN);
        lds_innorm[j1] = 1.0f / fmaxf(sqrtf(s1), EPSN);
    }
    __syncthreads();

    // ---- 4. dots via V_WMMA_F32_16X16X4_F32, 4 tiles of 16 refs, K swept 4 at a time
    // A layout (16x4 f32): lane<16 -> M=lane, holds K = k0+{0,1};
    //                      lane>=16 -> M=lane-16, holds K = k0+{2,3}.
    // B holds q replicated across all 16 N columns: per-lane value depends only
    // on the same K mapping, so every D column equals the desired dot vector.
    const int koff = (lane < 16) ? 0 : 2;     // select, no persistent divergence
    const int mrow = lane & 15;
    const int mhi  = (lane >= 16) ? 8 : 0;

    #pragma unroll
    for (int t = 0; t < 4; ++t) {
        const int       ridx = lds_idx[16 * t + mrow];
        const float*    rp   = r_embs + (size_t)ridx * DIM;
        v8f acc = {};
        #pragma unroll
        for (int k0 = 0; k0 < DIM; k0 += 4) {
            const v2f a  = *(const v2f*)(rp + k0 + koff);        // global_load_b64
            const v2f bb = *(const v2f*)(&lds_q[k0 + koff]);     // ds_load_b64
            acc = __builtin_amdgcn_wmma_f32_16x16x4_f32(
                      /*neg_a=*/false, a, /*neg_b=*/false, bb,
                      /*c_mod=*/(short)0, acc,
                      /*reuse_a=*/false, /*reuse_b=*/false);
        }
        // D layout: VGPR r, lanes 0-15 -> M=r, lanes 16-31 -> M=r+8; all N equal.
        #pragma unroll
        for (int r = 0; r < 8; ++r) {
            const int j = 16 * t + r + mhi;
            lds_dot[j] = acc[r] * inv_qn * lds_innorm[j] * INV_TEMP;
        }
    }
    __syncthreads();

    // ---- 5. logsumexp over 64 candidates + loss
    const float s0 = lds_dot[j0];
    const float s1 = lds_dot[j1];
    const float mx = wave_max32(fmaxf(s0, s1));
    const float se = wave_sum32(__expf(s0 - mx) + __expf(s1 - mx));
    if (lane == 0) {
        const float lse = mx + __logf(se);
        per_loss[b] = -lds_dot[0] + lse;
    }
}

// Deterministic single-workgroup mean reduction (fixed tree, no float atomics).
__global__ void __launch_bounds__(256)
mean_reduce_kernel(const float* __restrict__ per_loss, float* __restrict__ out, int n)
{
    __shared__ float sm[256];
    float s = 0.0f;
    for (int i = threadIdx.x; i < n; i += 256)
        s += per_loss[i];
    sm[threadIdx.x] = s;
    __syncthreads();
    #pragma unroll
    for (int w = 128; w > 0; w >>= 1) {
        if ((int)threadIdx.x < w) sm[threadIdx.x] += sm[threadIdx.x + w];
        __syncthreads();
    }
    if (threadIdx.x == 0)
        out[0] = sm[0] / (float)n;
}

extern "C" void kernel_launch(void* const* d_in, const int* in_sizes, int n_in,
                              void* d_out, int out_size, void* d_ws, size_t ws_size,
                              hipStream_t stream)
{
    const float* q_embs  = (const float*)d_in[0];   // [B,128] f32
    const float* r_embs  = (const float*)d_in[1];   // [V,128] f32
    const int*   pos_idx = (const int*)  d_in[2];   // [B] i32
    const int*   neg_idx = (const int*)  d_in[3];   // [B,63] i32

    const int batch = in_sizes[2];                  // 2048
    const int nneg  = in_sizes[3] / batch;          // 63

    float* per_loss = (float*)d_ws;                 // batch floats of scratch

    infonce_per_query_kernel<<<batch, 32, 0, stream>>>(
        q_embs, r_embs, pos_idx, neg_idx, per_loss, nneg);
    mean_reduce_kernel<<<1, 256, 0, stream>>>(per_loss, (float*)d_out, batch);
}